// SinkhornLoss_22703197127349
// MI455X (gfx1250) — compile-verified
//
#include <hip/hip_runtime.h>
#include <math.h>

// ---------------------------------------------------------------------------
// Sinkhorn loss on quaternion geodesic cost, MI455X (gfx1250).
//   N=4 batches, P1=P2=4096, d=4 (quaternions), eps=0.05, 50 iterations.
// Strategy:
//   * normalize x,y once
//   * per batch: materialize C (64 MB, L2-resident) with V_WMMA_F32_16X16X4_F32
//     (K=4 matches the quaternion dim), then 50x (col-LSE, row-LSE) sweeps
//   * LSE kernels run in base-2 (v_exp_f32 / v_log_f32 native) with two-phase
//     block logsumexp and b128 loads: ~5 VALU + 0.25 vmem per element
//   * final dual objective reduction -> d_out[0]
// ---------------------------------------------------------------------------

#define NB      4
#define PTS     4096
#define TILES   (PTS / 16)          // 256 tiles per dim
#define EPS_F   0.05f
#define CLIP_V  (1.0f - 1e-7f)
#define N_ITER  50

// base-2 constants
#define K2        28.853900817779268f   // (1/eps) * log2(e)
#define EPS_LN2   0.034657359027997264f // eps * ln2
#define EPS_LOGW  0.41588830833596715f  // -eps * log(1/4096)  (added at end)
#define M_INIT    (-1.0e30f)

typedef __attribute__((ext_vector_type(2))) float v2f;
typedef __attribute__((ext_vector_type(8))) float v8f;

__device__ __forceinline__ float exp2_fast(float x) {
    return __builtin_amdgcn_exp2f(x);
}
__device__ __forceinline__ float log2_fast(float x) {
    return __builtin_amdgcn_logf(x);
}
// merge (m,s) <- (m,s) ++ (m2,s2), base-2 logsumexp state
__device__ __forceinline__ void lse_merge(float& m, float& s, float m2, float s2) {
    float mx = fmaxf(m, m2);
    s = s * exp2_fast(m - mx) + s2 * exp2_fast(m2 - mx);
    m = mx;
}

// ---- normalize quaternions: out = q / |q| -----------------------------------
__global__ void sk_normalize(const float4* __restrict__ in,
                             float4* __restrict__ out, int n) {
    int t = blockIdx.x * blockDim.x + threadIdx.x;
    if (t < n) {
        float4 q = in[t];
        float r = rsqrtf(q.x * q.x + q.y * q.y + q.z * q.z + q.w * q.w);
        float4 o; o.x = q.x * r; o.y = q.y * r; o.z = q.z * r; o.w = q.w * r;
        out[t] = o;
    }
}

// ---- cost tiles: one wave computes one 16x16 tile via WMMA f32 16x16x4 ------
// A (16x4 f32): lane L holds M = L&15; VGPR v holds K = v + 2*(L>>4)
// B (4x16 f32): same striping with N = L&15
// D (16x16 f32, 8 VGPRs): VGPR r at lane L -> row M = r + 8*(L>>4), col N = L&15
__global__ void sk_cost_tiles(const float* __restrict__ xn,
                              const float* __restrict__ yn,
                              float* __restrict__ C) {
    int gw   = (blockIdx.x * blockDim.x + threadIdx.x) >> 5;  // global wave id
    int lane = threadIdx.x & 31;
    int ti = gw >> 8;          // row tile (0..255)
    int tj = gw & 255;         // col tile (0..255)
    int m  = lane & 15;
    int h  = lane >> 4;

    const float* xr = xn + (size_t)(ti * 16 + m) * 4;
    const float* yr = yn + (size_t)(tj * 16 + m) * 4;
    v2f a; a.x = xr[2 * h]; a.y = xr[2 * h + 1];
    v2f b; b.x = yr[2 * h]; b.y = yr[2 * h + 1];

    v8f c = {};
    c = __builtin_amdgcn_wmma_f32_16x16x4_f32(
            /*neg_a=*/false, a, /*neg_b=*/false, b,
            /*c_mod=*/(short)0, c, /*reuse_a=*/false, /*reuse_b=*/false);

    int col = tj * 16 + m;
#pragma unroll
    for (int r = 0; r < 8; ++r) {
        int row = ti * 16 + r + 8 * h;
        float d = fminf(fabsf(c[r]), CLIP_V);
        C[(size_t)row * PTS + col] = 2.0f * acosf(d);
    }
}

// ---- g update: g_j = -eps*( LSE_i[(f_i - C_ij)/eps] + log(1/P) ) ------------
// block = 256 threads = 16 col-groups(x4 cols, b128 loads) x 16 i-chunks(256 i)
__global__ void sk_g_update(const float* __restrict__ C,
                            const float* __restrict__ f,
                            float* __restrict__ g) {
    __shared__ float fs2[PTS];            // f[i] * K2
    __shared__ float ms[16][16][4];
    __shared__ float ss[16][16][4];

    int tx = threadIdx.x;
    for (int k = tx; k < PTS; k += 256) fs2[k] = f[k] * K2;
    __syncthreads();

    int jg    = tx & 15;                  // column group within block
    int chunk = tx >> 4;                  // i-chunk (16 chunks x 256 rows)
    int jq    = blockIdx.x * 16 + jg;     // float4 column index
    const float4* C4 = (const float4*)C;  // row stride PTS/4 float4s
    const int cs = PTS / 4;

    float m[4] = {M_INIT, M_INIT, M_INIT, M_INIT};
    float s[4] = {0.f, 0.f, 0.f, 0.f};
    int ibase = chunk * 256;

    for (int ib = 0; ib < 256; ib += 8) {
        float u[8][4];
#pragma unroll
        for (int r = 0; r < 8; ++r) {
            int i = ibase + ib + r;
            float4 cv = C4[(size_t)i * cs + jq];
            float fv = fs2[i];
            u[r][0] = fmaf(cv.x, -K2, fv);
            u[r][1] = fmaf(cv.y, -K2, fv);
            u[r][2] = fmaf(cv.z, -K2, fv);
            u[r][3] = fmaf(cv.w, -K2, fv);
        }
#pragma unroll
        for (int c = 0; c < 4; ++c) {
            float bm = u[0][c];
#pragma unroll
            for (int r = 1; r < 8; ++r) bm = fmaxf(bm, u[r][c]);
            float bs = 0.f;
#pragma unroll
            for (int r = 0; r < 8; ++r) bs += exp2_fast(u[r][c] - bm);
            lse_merge(m[c], s[c], bm, bs);
        }
    }
#pragma unroll
    for (int c = 0; c < 4; ++c) { ms[chunk][jg][c] = m[c]; ss[chunk][jg][c] = s[c]; }
    __syncthreads();

    if (tx < 64) {                        // one thread per output column
        int jg2 = tx >> 2, c = tx & 3;
        float mm = ms[0][jg2][c], sv = ss[0][jg2][c];
#pragma unroll
        for (int k = 1; k < 16; ++k) lse_merge(mm, sv, ms[k][jg2][c], ss[k][jg2][c]);
        g[blockIdx.x * 64 + tx] = -EPS_LN2 * (mm + log2_fast(sv)) + EPS_LOGW;
    }
}

// ---- f update: f_i = -eps*( LSE_j[(g_j - C_ij)/eps] + log(1/P) ) ------------
// one wave per row; per step each lane does 4 b128 loads (512 cols/wave/step)
__global__ void sk_f_update(const float* __restrict__ C,
                            const float* __restrict__ g,
                            float* __restrict__ f) {
    __shared__ float gs2[PTS];            // g[j] * K2
    int tx = threadIdx.x;
    for (int k = tx; k < PTS; k += 256) gs2[k] = g[k] * K2;
    __syncthreads();

    int row  = blockIdx.x * 8 + (tx >> 5);
    int lane = tx & 31;
    const float4* Crow4 = (const float4*)(C + (size_t)row * PTS);
    const float4* gs4   = (const float4*)gs2;

    float m = M_INIT, s = 0.f;
    for (int step = 0; step < 8; ++step) {
        int base = step * 128 + lane;     // float4 units
        float u[16];
#pragma unroll
        for (int q = 0; q < 4; ++q) {
            float4 cv = Crow4[base + 32 * q];
            float4 gv = gs4[base + 32 * q];
            u[4 * q + 0] = fmaf(cv.x, -K2, gv.x);
            u[4 * q + 1] = fmaf(cv.y, -K2, gv.y);
            u[4 * q + 2] = fmaf(cv.z, -K2, gv.z);
            u[4 * q + 3] = fmaf(cv.w, -K2, gv.w);
        }
        float bm = u[0];
#pragma unroll
        for (int k = 1; k < 16; ++k) bm = fmaxf(bm, u[k]);
        float bs = 0.f;
#pragma unroll
        for (int k = 0; k < 16; ++k) bs += exp2_fast(u[k] - bm);
        lse_merge(m, s, bm, bs);
    }
#pragma unroll
    for (int off = 16; off > 0; off >>= 1) {
        float m2 = __shfl_xor(m, off, 32);
        float s2 = __shfl_xor(s, off, 32);
        lse_merge(m, s, m2, s2);
    }
    if (lane == 0) f[row] = -EPS_LN2 * (m + log2_fast(s)) + EPS_LOGW;
}

// ---- final: out = sum_n [ mean(f_n) + mean(g_n) ] ---------------------------
__global__ void sk_final(const float* __restrict__ fb,
                         const float* __restrict__ gb,
                         float* __restrict__ out) {
    __shared__ float red[256];
    float acc = 0.0f;
    for (int k = threadIdx.x; k < NB * PTS; k += 256) acc += fb[k] + gb[k];
    red[threadIdx.x] = acc;
    __syncthreads();
    for (int s2 = 128; s2 > 0; s2 >>= 1) {
        if (threadIdx.x < s2) red[threadIdx.x] += red[threadIdx.x + s2];
        __syncthreads();
    }
    if (threadIdx.x == 0) out[0] = red[0] * (1.0f / PTS);
}

extern "C" void kernel_launch(void* const* d_in, const int* in_sizes, int n_in,
                              void* d_out, int out_size, void* d_ws, size_t ws_size,
                              hipStream_t stream) {
    const float* x_raw = (const float*)d_in[0];   // true_data (4,4096,4)
    const float* y_raw = (const float*)d_in[1];   // particles (4,4096,4)
    float* out = (float*)d_out;

    // workspace layout (~64.7 MB): C | xn | yn | f | g
    float* Cbuf = (float*)d_ws;                       // PTS*PTS
    float* xn   = Cbuf + (size_t)PTS * PTS;           // NB*PTS*4
    float* yn   = xn + (size_t)NB * PTS * 4;          // NB*PTS*4
    float* fbuf = yn + (size_t)NB * PTS * 4;          // NB*PTS
    float* gbuf = fbuf + (size_t)NB * PTS;            // NB*PTS
    (void)ws_size; (void)in_sizes; (void)n_in; (void)out_size;

    // normalize quaternions (both tensors)
    {
        int nq = NB * PTS;
        sk_normalize<<<(nq + 255) / 256, 256, 0, stream>>>(
            (const float4*)x_raw, (float4*)xn, nq);
        sk_normalize<<<(nq + 255) / 256, 256, 0, stream>>>(
            (const float4*)y_raw, (float4*)yn, nq);
    }

    // f0 = g0 = 0 (every call, for determinism)
    hipMemsetAsync(fbuf, 0, sizeof(float) * NB * PTS, stream);
    hipMemsetAsync(gbuf, 0, sizeof(float) * NB * PTS, stream);

    // batches sequentially so the 64 MB C stays L2-resident across 100 sweeps
    for (int n = 0; n < NB; ++n) {
        const float* xb = xn + (size_t)n * PTS * 4;
        const float* yb = yn + (size_t)n * PTS * 4;
        float* fb = fbuf + (size_t)n * PTS;
        float* gb = gbuf + (size_t)n * PTS;

        // 256x256 tiles, 1 wave each, 8 waves/block -> 8192 blocks
        sk_cost_tiles<<<(TILES * TILES) / 8, 256, 0, stream>>>(xb, yb, Cbuf);

        for (int it = 0; it < N_ITER; ++it) {
            sk_g_update<<<PTS / 64, 256, 0, stream>>>(Cbuf, fb, gb);
            sk_f_update<<<PTS / 8, 256, 0, stream>>>(Cbuf, gb, fb);
        }
    }

    sk_final<<<1, 256, 0, stream>>>(fbuf, gbuf, out);
}